// HeteroGraphSAGE_11785390260819
// MI455X (gfx1250) — compile-verified
//
#include <hip/hip_runtime.h>

typedef float v2f __attribute__((ext_vector_type(2)));
typedef float v8f __attribute__((ext_vector_type(8)));

constexpr int NU = 200000, NP = 50000, NL = 100000;
constexpr int EU = 1000000, EV = 1000000;
constexpr int H = 64;

// ---------------- input projection: y[n][64] = x[n][K] @ W[K][64] + b ----------------
template <int K>
__global__ void proj_kernel(const float* __restrict__ x, const float* __restrict__ W,
                            const float* __restrict__ b, float* __restrict__ y, int n) {
  int i = blockIdx.x * blockDim.x + threadIdx.x;
  if (i >= n * H) return;
  int node = i >> 6, c = i & 63;
  float acc = b[c];
#pragma unroll
  for (int k = 0; k < K; ++k) acc = fmaf(x[node * K + k], W[k * H + c], acc);
  y[i] = acc;
}

// ---------------- per-destination degree count ----------------
__global__ void count_kernel(const int* __restrict__ dst, float* cnt, int E) {
  int e = blockIdx.x * blockDim.x + threadIdx.x;
  if (e < E) atomicAdd(&cnt[dst[e]], 1.0f);
}

// ---------------- edge scatter: agg[dst] += feat[src], 2 cols per thread ----------------
__global__ void scatter_kernel(const int* __restrict__ src, const int* __restrict__ dst,
                               const float* __restrict__ feat, float* agg, int E) {
  int i = blockIdx.x * blockDim.x + threadIdx.x;  // over E*32
  if (i >= E * 32) return;
  int e = i >> 5, c = (i & 31) << 1;
  int s = src[e], d = dst[e];
  const float2 v = *reinterpret_cast<const float2*>(&feat[(size_t)s * H + c]);
  atomicAdd(&agg[(size_t)d * H + c], v.x);
  atomicAdd(&agg[(size_t)d * H + c + 1], v.y);
}

// ---------------- fused SAGE node update via fp32 WMMA ----------------
// out[n][64] = relu( sum_t scale_t(n) * A_t[n][64] @ W_t[64][64]  + bias0 + bias1 )
// scale_t = 1/cnt (0 if cnt==0) when cnt ptr given, else 1.  Up to 4 terms.
// One wave computes a 16x64 output tile; K loop: 16 steps of V_WMMA_F32_16X16X4_F32.
__global__ __launch_bounds__(64) void sage_update_kernel(
    float* out,
    const float* A0, const float* c0, const float* W0,
    const float* A1, const float* c1, const float* W1,
    const float* A2, const float* c2, const float* W2,
    const float* A3, const float* c3, const float* W3,
    const float* bias0, const float* bias1, int n_nodes) {
  const int lane = threadIdx.x & 31;
  const int wave = (blockIdx.x * blockDim.x + threadIdx.x) >> 5;
  const int rowBase = wave * 16;
  if (rowBase >= n_nodes) return;      // uniform per wave; EXEC stays all-1 for WMMA
  const int half = lane >> 4;          // lane group (0: lanes 0-15, 1: lanes 16-31)
  const int l16 = lane & 15;
  const int row = rowBase + l16;       // A-matrix row owned by this lane

  v8f acc0 = {0, 0, 0, 0, 0, 0, 0, 0};
  v8f acc1 = acc0, acc2 = acc0, acc3 = acc0;

  const float* As[4] = {A0, A1, A2, A3};
  const float* Cs[4] = {c0, c1, c2, c3};
  const float* Ws[4] = {W0, W1, W2, W3};

#pragma unroll
  for (int t = 0; t < 4; ++t) {
    const float* A = As[t];
    if (!A) continue;
    const float* Wm = Ws[t];
    float scale = 1.0f;
    if (Cs[t]) {
      float cv = Cs[t][row];
      scale = cv > 0.f ? 1.0f / cv : 0.f;
    }
#pragma unroll
    for (int kt = 0; kt < 16; ++kt) {
      const int k0 = kt * 4 + half * 2;
      // A 16x4 fragment: lane holds A[row][k0], A[row][k0+1]
      v2f a;
      a.x = A[(size_t)row * H + k0] * scale;
      a.y = A[(size_t)row * H + k0 + 1] * scale;
      // B 4x16 fragments (one per 16-col output tile): lane holds W[k0..k0+1][col]
      v2f b0, b1, b2, b3;
      b0.x = Wm[(k0) * H + 0 + l16];  b0.y = Wm[(k0 + 1) * H + 0 + l16];
      b1.x = Wm[(k0) * H + 16 + l16]; b1.y = Wm[(k0 + 1) * H + 16 + l16];
      b2.x = Wm[(k0) * H + 32 + l16]; b2.y = Wm[(k0 + 1) * H + 32 + l16];
      b3.x = Wm[(k0) * H + 48 + l16]; b3.y = Wm[(k0 + 1) * H + 48 + l16];
      acc0 = __builtin_amdgcn_wmma_f32_16x16x4_f32(false, a, false, b0, (short)0, acc0, false, false);
      acc1 = __builtin_amdgcn_wmma_f32_16x16x4_f32(false, a, false, b1, (short)0, acc1, false, false);
      acc2 = __builtin_amdgcn_wmma_f32_16x16x4_f32(false, a, false, b2, (short)0, acc2, false, false);
      acc3 = __builtin_amdgcn_wmma_f32_16x16x4_f32(false, a, false, b3, (short)0, acc3, false, false);
    }
  }

  // bias per output column
  float bv0 = 0.f, bv1 = 0.f, bv2 = 0.f, bv3 = 0.f;
  if (bias0) {
    bv0 += bias0[0 + l16]; bv1 += bias0[16 + l16];
    bv2 += bias0[32 + l16]; bv3 += bias0[48 + l16];
  }
  if (bias1) {
    bv0 += bias1[0 + l16]; bv1 += bias1[16 + l16];
    bv2 += bias1[32 + l16]; bv3 += bias1[48 + l16];
  }

  // C/D layout: VGPR r -> row (r + 8*half), col l16 (+16*ntile)
#pragma unroll
  for (int r = 0; r < 8; ++r) {
    const size_t gr = (size_t)(rowBase + r + 8 * half) * H;
    out[gr + 0 + l16]  = fmaxf(acc0[r] + bv0, 0.f);
    out[gr + 16 + l16] = fmaxf(acc1[r] + bv1, 0.f);
    out[gr + 32 + l16] = fmaxf(acc2[r] + bv2, 0.f);
    out[gr + 48 + l16] = fmaxf(acc3[r] + bv3, 0.f);
  }
}

// ---------------- classifier: out = relu(hu@cW1+cb1) @ cW2 + cb2 ----------------
__global__ void classifier_kernel(const float* __restrict__ hu, const float* __restrict__ cW1,
                                  const float* __restrict__ cb1, const float* __restrict__ cW2,
                                  const float* __restrict__ cb2, float* __restrict__ out, int n) {
  int node = blockIdx.x * blockDim.x + threadIdx.x;
  if (node >= n) return;
  const float* x = hu + (size_t)node * H;
  float h[32];
  for (int j = 0; j < 32; ++j) {
    float acc = cb1[j];
#pragma unroll 8
    for (int k = 0; k < H; ++k) acc = fmaf(x[k], cW1[k * 32 + j], acc);
    h[j] = fmaxf(acc, 0.f);
  }
  float o0 = cb2[0], o1 = cb2[1];
#pragma unroll
  for (int j = 0; j < 32; ++j) {
    o0 = fmaf(h[j], cW2[j * 2 + 0], o0);
    o1 = fmaf(h[j], cW2[j * 2 + 1], o1);
  }
  out[(size_t)node * 2 + 0] = o0;
  out[(size_t)node * 2 + 1] = o1;
}

extern "C" void kernel_launch(void* const* d_in, const int* in_sizes, int n_in,
                              void* d_out, int out_size, void* d_ws, size_t ws_size,
                              hipStream_t stream) {
  (void)in_sizes; (void)n_in; (void)out_size; (void)ws_size;
  const float* x_user = (const float*)d_in[0];
  const float* x_pc   = (const float*)d_in[1];
  const float* x_url  = (const float*)d_in[2];
  const int* edge_uses   = (const int*)d_in[3];
  const int* edge_visits = (const int*)d_in[4];
  const float* Wu = (const float*)d_in[5];
  const float* bu = (const float*)d_in[6];
  const float* Wp = (const float*)d_in[7];
  const float* bp = (const float*)d_in[8];
  const float* Wl = (const float*)d_in[9];
  const float* bl = (const float*)d_in[10];
  const float* sage_l_W = (const float*)d_in[11];
  const float* sage_l_b = (const float*)d_in[12];
  const float* sage_r_W = (const float*)d_in[13];
  const float* cW1 = (const float*)d_in[14];
  const float* cb1 = (const float*)d_in[15];
  const float* cW2 = (const float*)d_in[16];
  const float* cb2 = (const float*)d_in[17];

  const int* uses_src = edge_uses;           // user ids
  const int* uses_dst = edge_uses + EU;      // pc ids
  const int* visits_src = edge_visits;       // user ids
  const int* visits_dst = edge_visits + EV;  // url ids

  // ---- workspace layout (floats) ----
  float* ws = (float*)d_ws;
  size_t off = 0;
  float* hu = ws + off; off += (size_t)NU * H;
  float* hp = ws + off; off += (size_t)NP * H;
  float* hl = ws + off; off += (size_t)NL * H;
  float* agg_upc  = ws + off; off += (size_t)NU * H;  // contiguous agg block start
  float* agg_uurl = ws + off; off += (size_t)NU * H;
  float* agg_pc   = ws + off; off += (size_t)NP * H;
  float* agg_url  = ws + off; off += (size_t)NL * H;
  float* cnt_upc  = ws + off; off += NU;              // contiguous count block start
  float* cnt_uurl = ws + off; off += NU;
  float* cnt_pc   = ws + off; off += NP;
  float* cnt_url  = ws + off; off += NL;

  // ---- degree counts (graph is fixed; recomputed each call for determinism) ----
  hipMemsetAsync(cnt_upc, 0, sizeof(float) * (size_t)(2 * NU + NP + NL), stream);
  count_kernel<<<(EU + 255) / 256, 256, 0, stream>>>(uses_dst, cnt_pc, EU);
  count_kernel<<<(EU + 255) / 256, 256, 0, stream>>>(uses_src, cnt_upc, EU);
  count_kernel<<<(EV + 255) / 256, 256, 0, stream>>>(visits_dst, cnt_url, EV);
  count_kernel<<<(EV + 255) / 256, 256, 0, stream>>>(visits_src, cnt_uurl, EV);

  // ---- input projections ----
  proj_kernel<6><<<((size_t)NU * H + 255) / 256, 256, 0, stream>>>(x_user, Wu, bu, hu, NU);
  proj_kernel<4><<<((size_t)NP * H + 255) / 256, 256, 0, stream>>>(x_pc, Wp, bp, hp, NP);
  proj_kernel<3><<<((size_t)NL * H + 255) / 256, 256, 0, stream>>>(x_url, Wl, bl, hl, NL);

  for (int layer = 0; layer < 2; ++layer) {
    const float* WLb = sage_l_W + (size_t)layer * 4 * H * H;
    const float* BLb = sage_l_b + (size_t)layer * 4 * H;
    const float* WRb = sage_r_W + (size_t)layer * 4 * H * H;
#define WLr(r) (WLb + (size_t)(r) * H * H)
#define BLr(r) (BLb + (size_t)(r) * H)
#define WRr(r) (WRb + (size_t)(r) * H * H)

    // zero all accumulators (one contiguous memset)
    hipMemsetAsync(agg_upc, 0, sizeof(float) * (size_t)(2 * NU + NP + NL) * H, stream);

    // rel0: user-uses->pc   : agg_pc[dst=pc]  += hu[src=user]
    scatter_kernel<<<((size_t)EU * 32 + 255) / 256, 256, 0, stream>>>(uses_src, uses_dst, hu, agg_pc, EU);
    // rel1: pc-used_by->user: agg_upc[user]   += hp[pc]
    scatter_kernel<<<((size_t)EU * 32 + 255) / 256, 256, 0, stream>>>(uses_dst, uses_src, hp, agg_upc, EU);
    // rel2: user-visits->url: agg_url[url]    += hu[user]
    scatter_kernel<<<((size_t)EV * 32 + 255) / 256, 256, 0, stream>>>(visits_src, visits_dst, hu, agg_url, EV);
    // rel3: url-visited_by->user: agg_uurl[user] += hl[url]
    scatter_kernel<<<((size_t)EV * 32 + 255) / 256, 256, 0, stream>>>(visits_dst, visits_src, hl, agg_uurl, EV);

    // pc:  relu(mean_pc@WL0 + BL0 + hp@WR0)   (in-place on hp: wave reads own rows first)
    sage_update_kernel<<<(NP / 16 + 1) / 2, 64, 0, stream>>>(
        hp, agg_pc, cnt_pc, WLr(0), hp, nullptr, WRr(0),
        nullptr, nullptr, nullptr, nullptr, nullptr, nullptr,
        BLr(0), nullptr, NP);
    // url: relu(mean_url@WL2 + BL2 + hl@WR2)
    sage_update_kernel<<<(NL / 16 + 1) / 2, 64, 0, stream>>>(
        hl, agg_url, cnt_url, WLr(2), hl, nullptr, WRr(2),
        nullptr, nullptr, nullptr, nullptr, nullptr, nullptr,
        BLr(2), nullptr, NL);
    // user: relu(mean_upc@WL1 + BL1 + mean_uurl@WL3 + BL3 + hu@WR1 + hu@WR3)
    sage_update_kernel<<<(NU / 16 + 1) / 2, 64, 0, stream>>>(
        hu, agg_upc, cnt_upc, WLr(1), agg_uurl, cnt_uurl, WLr(3),
        hu, nullptr, WRr(1), hu, nullptr, WRr(3),
        BLr(1), BLr(3), NU);
#undef WLr
#undef BLr
#undef WRr
  }

  classifier_kernel<<<(NU + 255) / 256, 256, 0, stream>>>(hu, cW1, cb1, cW2, cb2, (float*)d_out, NU);
}